// VoxelResBackBone8x_SA_52570399703403
// MI455X (gfx1250) — compile-verified
//
#include <hip/hip_runtime.h>
#include <math.h>

// ---------------- problem constants (from the reference) ----------------
constexpr int B_    = 2;
constexpr int N_PER = 16384;
constexpr int M_PER = 8192;
constexpr int C_    = 64;
constexpr int NS_   = 16;
constexpr int ZD = 10, YD = 400, XD = 352;
constexpr float R2 = 1.2f * 1.2f;
constexpr int WAVE = 32;   // gfx1250 is wave32-only

typedef __attribute__((ext_vector_type(2))) float v2f;
typedef __attribute__((ext_vector_type(8))) float v8f;

// ---------------- kernel 1: voxel centers (+ precomputed |p|^2 in .w) ----
__global__ void centers_kernel(const int* __restrict__ coords,
                               float4* __restrict__ out, int n) {
  int i = blockIdx.x * blockDim.x + threadIdx.x;
  if (i >= n) return;
  int z = coords[i * 4 + 1];
  int y = coords[i * 4 + 2];
  int x = coords[i * 4 + 3];
  // VOXEL * DS = (0.2, 0.2, 0.4); PC_MIN = (0, -40, -3)
  float fx = ((float)x + 0.5f) * 0.2f + 0.0f;
  float fy = ((float)y + 0.5f) * 0.2f - 40.0f;
  float fz = ((float)z + 0.5f) * 0.4f - 3.0f;
  out[i] = make_float4(fx, fy, fz, fx * fx + fy * fy + fz * fz);
}

// ---------------- kernel 2: WMMA ball grouping (16 queries per wave) ----
// d2(q,p) = |q|^2 + |p|^2 - 2 q.p is a rank-4 bilinear form:
//   A row (query m)  = (-2qx, -2qy, -2qz, 1)        (16x4)
//   B col (point n)  = ( px,   py,   pz,  |p|^2)    (4x16)
//   C[m][n]          = |p|^2 - 2 q.p   -> within iff C < R2 - |q|^2
// Two V_WMMA_F32_16X16X4_F32 per iteration score a 16x32 (query x point)
// slab; the two 16-bit per-query masks concatenate into one 32-bit mask in
// ascending point-index order (LSB = lowest index), which matches the
// reference's top_k(score = -index) selection exactly.  Point tiles are
// software-pipelined: next slab's loads issue before the current slab's
// math (only ~1 wave/SIMD here, so no multi-wave latency hiding).
__global__ void __launch_bounds__(128)
group_kernel(const float4* __restrict__ xyz,
             const float4* __restrict__ nxyz,
             int* __restrict__ idxOut) {
  __shared__ int sIdx[4][256];   // per-wave: 16 queries x 16 slots
  __shared__ int sCnt[4][16];

  const int wave = threadIdx.x / WAVE;
  const int lane = threadIdx.x % WAVE;
  const int s    = lane & 15;
  const int half = lane >> 4;
  const int q0 = (blockIdx.x * 4 + wave) * 16;   // 16 queries per wave
  const int b  = q0 / M_PER;                     // slabs never straddle batches
  const float4* pts = xyz + b * N_PER;

  // constant A fragment: lane holds query s, cols (half*2, half*2+1)
  float4 qc = nxyz[q0 + s];
  v2f a;
  a.x = half ? (-2.0f * qc.z) : (-2.0f * qc.x);
  a.y = half ? 1.0f           : (-2.0f * qc.y);

  // per-C-register thresholds: thr[r] = R2 - |q_{r+half*8}|^2
  float thr[8];
#pragma unroll
  for (int r = 0; r < 8; ++r) thr[r] = R2 - __shfl(qc.w, r + half * 8, WAVE);

  int* slots = &sIdx[wave][0];
  int cnt = (lane < 16) ? 0 : NS_;   // lanes >=16 never append / never stall exit

  // pipelined first slab
  float4 p0 = pts[s];
  float4 p1 = pts[16 + s];

  for (int base = 0; base < N_PER; base += 32) {
    // issue next slab's loads before this slab's math
    int nb = (base + 32 < N_PER) ? (base + 32) : 0;
    float4 n0 = pts[nb + s];
    float4 n1 = pts[nb + 16 + s];

    v2f bf0, bf1;
    bf0.x = half ? p0.z : p0.x;  bf0.y = half ? p0.w : p0.y;
    bf1.x = half ? p1.z : p1.x;  bf1.y = half ? p1.w : p1.y;

    v8f z8 = {0.f, 0.f, 0.f, 0.f, 0.f, 0.f, 0.f, 0.f};
    v8f acc0 = __builtin_amdgcn_wmma_f32_16x16x4_f32(false, a, false, bf0,
                                                     (short)0, z8, false, false);
    v8f acc1 = __builtin_amdgcn_wmma_f32_16x16x4_f32(false, a, false, bf1,
                                                     (short)0, z8, false, false);

    // compares double as ballots: 16 query masks (16 bits each, LSB = low idx)
    unsigned bal0[8], bal1[8];
#pragma unroll
    for (int r = 0; r < 8; ++r) {
      bal0[r] = (unsigned)__ballot(acc0[r] < thr[r]);
      bal1[r] = (unsigned)__ballot(acc1[r] < thr[r]);
    }
    // route query (lane)'s masks to lanes 0..15: m<8 -> bal[m] lo16,
    // m>=8 -> bal[m-8] hi16.  bal[] uniform -> cndmask chains.
    unsigned u0 = bal0[0], u1 = bal1[0];
#pragma unroll
    for (int r = 1; r < 8; ++r) {
      bool sel = (lane & 7) == r;
      u0 = sel ? bal0[r] : u0;
      u1 = sel ? bal1[r] : u1;
    }
    unsigned lo = (lane < 8) ? (u0 & 0xffffu) : (u0 >> 16);
    unsigned hi = (lane < 8) ? (u1 & 0xffffu) : (u1 >> 16);
    unsigned mask = (lane < 16) ? (lo | (hi << 16)) : 0u;

    // each of lanes 0..15 drains its own query's hits (usually empty)
    while (mask && cnt < NS_) {
      int bit = __ffs(mask) - 1;
      slots[lane * 16 + cnt] = base + bit;
      cnt++;
      mask &= mask - 1;
    }
    p0 = n0;
    p1 = n1;
    if ((unsigned)__ballot(cnt < NS_) == 0u) break;   // all 16 queries full
  }
  if (lane < 16) sCnt[wave][lane] = cnt;
  __syncthreads();   // cross-lane LDS handoff (uniform across the block)

  // finalize: duplicate slot0 into unfilled slots, -1 marks empty group
#pragma unroll
  for (int e = 0; e < 256; e += WAVE) {
    int idx = e + lane;
    int qm = idx >> 4, sl = idx & 15;
    int cq = sCnt[wave][qm];
    int v = (sl < cq) ? slots[qm * 16 + sl]
                      : (cq > 0 ? slots[qm * 16] : -1);
    idxOut[(q0 + qm) * NS_ + sl] = v;
  }
}

// ---------------- kernel 3: gather + MLP(64->64->64) + maxpool ----------
// One wave per query.  All matmuls via V_WMMA_F32_16X16X4_F32 (fp32).
// LDS in WMMA fragment order: every fragment fetch is one aligned,
// bank-conflict-free ds_load_b64 into an even VGPR pair.
constexpr int FRAG = 64;        // dwords per weight fragment (32 lanes x 2)
constexpr int SFRAG = 66;       // padded scratch fragment stride (conflict-free)

__global__ void __launch_bounds__(128)
mlp_kernel(const float* __restrict__ features,
           const int* __restrict__ idxBuf,
           const float* __restrict__ W1, const float* __restrict__ g1,
           const float* __restrict__ b1,
           const float* __restrict__ W2, const float* __restrict__ g2,
           const float* __restrict__ b2,
           float* __restrict__ pooled) {
  __shared__ float sW1[64 * FRAG];           // 4096 dwords = 16 KB
  __shared__ float sW2[64 * FRAG];
  __shared__ float sH[4][16 * SFRAG];        // per-wave h1 scratch (frag order)

  // cooperative weight staging into fragment order:
  // W[k][n] -> frag(k>>2, n>>4), lane ((k&3)>>1)*16+(n&15), comp k&1
  for (int i = threadIdx.x; i < 64 * 64; i += blockDim.x) {
    int k = i >> 6, n = i & 63;
    int kc = k >> 2, sub = k & 3;
    int ln = ((sub >> 1) << 4) | (n & 15);
    int pos = (kc * 4 + (n >> 4)) * FRAG + ln * 2 + (sub & 1);
    sW1[pos] = W1[i];
    sW2[pos] = W2[i];
  }
  __syncthreads();

  const int wave = threadIdx.x / WAVE;
  const int lane = threadIdx.x % WAVE;
  const int q = blockIdx.x * 4 + wave;
  const int s = lane & 15;            // row within 16x16 tile
  const int half = lane >> 4;         // which half-wave
  const int koff = half * 2;          // K sub-offset for A fragments
  float* scr = &sH[wave][0];

  // gather layer-1 A fragments: row = features[idx], -1 -> zero row
  int row = idxBuf[q * NS_ + s];
  const float* frow = (row >= 0) ? (features + (size_t)row * C_) : nullptr;
  v2f a1[16];
#pragma unroll
  for (int kc = 0; kc < 16; ++kc) {
    int c0 = kc * 4 + koff;
    if (frow) { a1[kc] = *(const v2f*)(frow + c0); }
    else      { a1[kc].x = 0.0f; a1[kc].y = 0.0f; }
  }

  // ---- layer 1: 4 N-tiles x 16 K-steps of wmma_f32_16x16x4_f32 ----
#pragma unroll
  for (int t = 0; t < 4; ++t) {
    v8f acc = {0.f, 0.f, 0.f, 0.f, 0.f, 0.f, 0.f, 0.f};
#pragma unroll
    for (int kc = 0; kc < 16; ++kc) {
      v2f bf = *(const v2f*)&sW1[(kc * 4 + t) * FRAG + lane * 2];
      acc = __builtin_amdgcn_wmma_f32_16x16x4_f32(false, a1[kc], false, bf,
                                                  (short)0, acc, false, false);
    }
    // relu(acc*g1+b1), scatter-store into A-fragment order for layer 2.
    // C layout: reg r = (m = r + half*8, col = t*16 + s).
    int col = t * 16 + s;
    float sg = g1[col], sb = b1[col];
    int kc2  = col >> 2;                      // destination fragment
    int jj   = col & 1;                       // component within b64
    int lbase = ((col & 2) << 3) + half * 8;  // ((col&3)>>1)*16 + half*8
    int abase = kc2 * SFRAG + lbase * 2 + jj;
#pragma unroll
    for (int r = 0; r < 8; ++r) {
      float v = acc[r] * sg + sb;
      scr[abase + r * 2] = v > 0.0f ? v : 0.0f;
    }
  }
  __syncthreads();   // cross-lane LDS scratch handoff

  // reload h1 as layer-2 A fragments: one b64 per k-chunk
  v2f a2[16];
#pragma unroll
  for (int kc = 0; kc < 16; ++kc) {
    a2[kc] = *(const v2f*)&scr[kc * SFRAG + lane * 2];
  }

  // ---- layer 2 + max-pool over the 16 neighbor rows ----
  float* pq = pooled + (size_t)q * C_;
#pragma unroll
  for (int t = 0; t < 4; ++t) {
    v8f acc = {0.f, 0.f, 0.f, 0.f, 0.f, 0.f, 0.f, 0.f};
#pragma unroll
    for (int kc = 0; kc < 16; ++kc) {
      v2f bf = *(const v2f*)&sW2[(kc * 4 + t) * FRAG + lane * 2];
      acc = __builtin_amdgcn_wmma_f32_16x16x4_f32(false, a2[kc], false, bf,
                                                  (short)0, acc, false, false);
    }
    int col = t * 16 + s;
    float sg = g2[col], sb = b2[col];
    float mx = 0.0f;                  // relu output >= 0, so 0 is identity
#pragma unroll
    for (int r = 0; r < 8; ++r) {
      float v = acc[r] * sg + sb;
      mx = fmaxf(mx, v);              // fmax over relu'd rows
    }
    mx = fmaxf(mx, __shfl_xor(mx, 16, WAVE));  // combine M=0..7 with M=8..15
    if (lane < 16) pq[col] = mx;
  }
}

// ---------------- kernel 4a: key -> sub-index lookup table --------------
__global__ void scatter_kernel(const int* __restrict__ sub_coords,
                               int* __restrict__ table) {
  int j = blockIdx.x * blockDim.x + threadIdx.x;
  if (j >= B_ * M_PER) return;
  int b = sub_coords[j * 4 + 0];
  int z = sub_coords[j * 4 + 1];
  int y = sub_coords[j * 4 + 2];
  int x = sub_coords[j * 4 + 3];
  int key = ((b * ZD + z) * YD + y) * XD + x;
  table[key] = j;
}

// ---------------- kernel 4b: fused add + BN-affine + residual relu ------
__global__ void fuse_kernel(const float* __restrict__ features,
                            const float* __restrict__ pooled,
                            const int* __restrict__ coords,
                            const int* __restrict__ table,
                            const float* __restrict__ bn_g,
                            const float* __restrict__ bn_b,
                            float* __restrict__ out) {
  int tid = blockIdx.x * blockDim.x + threadIdx.x;  // (point, float4-chunk)
  int i = tid >> 4;
  int c4 = tid & 15;
  if (i >= B_ * N_PER) return;
  int b = coords[i * 4 + 0];
  int z = coords[i * 4 + 1];
  int y = coords[i * 4 + 2];
  int x = coords[i * 4 + 3];
  int key = ((b * ZD + z) * YD + y) * XD + x;
  int j = table[key];

  float4 f = ((const float4*)(features + (size_t)i * C_))[c4];
  float4 add = make_float4(0.f, 0.f, 0.f, 0.f);
  if (j >= 0) add = ((const float4*)(pooled + (size_t)j * C_))[c4];
  float4 g  = ((const float4*)bn_g)[c4];
  float4 bb = ((const float4*)bn_b)[c4];
  float4 r;
  r.x = fmaxf((f.x + add.x) * g.x + bb.x + f.x, 0.0f);
  r.y = fmaxf((f.y + add.y) * g.y + bb.y + f.y, 0.0f);
  r.z = fmaxf((f.z + add.z) * g.z + bb.z + f.z, 0.0f);
  r.w = fmaxf((f.w + add.w) * g.w + bb.w + f.w, 0.0f);
  ((float4*)out)[tid] = r;
}

// ---------------- launch --------------------------------------------------
extern "C" void kernel_launch(void* const* d_in, const int* in_sizes, int n_in,
                              void* d_out, int out_size, void* d_ws, size_t ws_size,
                              hipStream_t stream) {
  const float* features = (const float*)d_in[0];
  const float* W1   = (const float*)d_in[1];
  const float* g1   = (const float*)d_in[2];
  const float* b1   = (const float*)d_in[3];
  const float* W2   = (const float*)d_in[4];
  const float* g2   = (const float*)d_in[5];
  const float* b2   = (const float*)d_in[6];
  const float* bn2g = (const float*)d_in[7];
  const float* bn2b = (const float*)d_in[8];
  const int* coords     = (const int*)d_in[9];
  const int* sub_coords = (const int*)d_in[10];
  float* out = (float*)d_out;

  // workspace layout (bytes):
  //   xyz     :  524288   (32768 x float4, |p|^2 in .w)
  //   nxyz    :  262144   (16384 x float4)
  //   idx     : 1048576   (16384 x 16 int)
  //   pooled  : 4194304   (16384 x 64 float)
  //   table   : 11264000  (B*Z*Y*X int, -1 = no match) -> total ~17.3 MB
  char* ws = (char*)d_ws;
  float4* xyz   = (float4*)(ws);
  float4* nxyz  = (float4*)(ws + 524288);
  int*    idx   = (int*)  (ws + 786432);
  float*  pool  = (float*)(ws + 1835008);
  int*    table = (int*)  (ws + 6029312);

  centers_kernel<<<(B_ * N_PER + 255) / 256, 256, 0, stream>>>(coords, xyz, B_ * N_PER);
  centers_kernel<<<(B_ * M_PER + 255) / 256, 256, 0, stream>>>(sub_coords, nxyz, B_ * M_PER);
  hipMemsetAsync(table, 0xFF, (size_t)B_ * ZD * YD * XD * sizeof(int), stream);

  // 16 queries per wave, 4 waves per block -> 256 blocks
  group_kernel<<<(B_ * M_PER) / 64, 128, 0, stream>>>(xyz, nxyz, idx);
  mlp_kernel<<<(B_ * M_PER) / 4, 128, 0, stream>>>(features, idx, W1, g1, b1,
                                                   W2, g2, b2, pool);
  scatter_kernel<<<(B_ * M_PER + 255) / 256, 256, 0, stream>>>(sub_coords, table);
  fuse_kernel<<<(B_ * N_PER * 16 + 255) / 256, 256, 0, stream>>>(
      features, pool, coords, table, bn2g, bn2b, out);
}